// FeatureVolume_65506841199154
// MI455X (gfx1250) — compile-verified
//
#include <hip/hip_runtime.h>
#include <hip/hip_bf16.h>
#include <cstdint>

#define FV_S      129
#define FV_C      32
#define FV_V      (129LL * 129LL * 129LL)   /* 2,146,689 voxels */
#define FV_PPW    8                          /* points per wave */

// Primary: TDM via inline asm (portable across toolchains per bridge doc).
// Set to 0 to fall back to the per-lane async global->LDS path.
#define FV_USE_TDM_ASM 1

typedef unsigned int fv_v4u __attribute__((ext_vector_type(4)));
typedef unsigned int fv_v8u __attribute__((ext_vector_type(8)));

// ---------------------------------------------------------------------------
// Kernel 1: layout transform  fm[C][V]  ->  fmT[V][C]
// Phase 1 stages a 256-voxel x 32-channel tile in LDS:
//   * TDM path: one tensor_load_to_lds per workgroup (wave 0 issues; the DMA
//     engine streams the 2D tile with row stride V and inserts 1 DWORD of LDS
//     padding every 256 DWORDs -> LDS row stride 257 floats, conflict-free).
//   * Fallback: per-lane global_load_async_to_lds_b32 (ASYNCcnt).
// Phase 2 writes the tile out fully coalesced in [V][C] order.
// ---------------------------------------------------------------------------
__global__ __launch_bounds__(256) void fv_transpose(const float* __restrict__ fm,
                                                    float* __restrict__ fmT) {
    __shared__ float tile[8448];            // >= max(32*257, 256*33) floats
    const int  t  = threadIdx.x;
    const long long v0 = (long long)blockIdx.x * 256;
    const long long v  = v0 + t;

#if FV_USE_TDM_ASM
    if (threadIdx.x < 32) {                 // one wave issues the DMA (TDM ignores EXEC)
        const unsigned lds_base = (unsigned)(uintptr_t)(&tile[0]);
        const unsigned long long gaddr =
            (unsigned long long)(uintptr_t)fm + (unsigned long long)v0 * 4ull;
        const unsigned td0 = (unsigned)(FV_V - v0);   // remaining voxels: OOB reads -> 0

        // D# group 0 (ISA 8.3): count=1 | lds_addr | global_addr[56:0] | type=2
        fv_v4u g0;
        g0.x = 1u;
        g0.y = lds_base;
        g0.z = (unsigned)(gaddr & 0xFFFFFFFFull);
        g0.w = (unsigned)((gaddr >> 32) & 0x1FFFFFFull) | (2u << 30);

        // D# group 1 (ISA 8.4)
        fv_v8u g1;
        g1[0] = 0x01D20000u;                 // data_size=4B | pad_enable | interval=256dw | amount=1dw
        g1[1] = (td0 & 0xFFFFu) << 16;                    // tensor_dim0[15:0]
        g1[2] = ((td0 >> 16) & 0xFFFFu) | (32u << 16);    // tensor_dim0[31:16] | tensor_dim1=32
        g1[3] = 256u << 16;                  // tile_dim0 = 256
        g1[4] = 32u;                         // tile_dim1 = 32 (tile_dim2 = 0)
        g1[5] = 2146689u;                    // tensor_dim0_stride = V (0x20C181)
        g1[6] = 0u;
        g1[7] = 0u;

        // 2D tensor: groups 2/3 are NULL -> 2-operand assembler form.
        asm volatile("tensor_load_to_lds %0, %1\n\t"
                     "s_wait_tensorcnt 0x0"
                     :: "s"(g0), "s"(g1) : "memory");
    }
    __syncthreads();
#else
    if (v < FV_V) {
#pragma unroll
        for (int c = 0; c < FV_C; ++c) {
            const float* gp  = fm + (long long)c * FV_V + v;
            unsigned lds_off = (unsigned)(uintptr_t)(&tile[t * 33 + c]);
            asm volatile("global_load_async_to_lds_b32 %0, %1, off"
                         :: "v"(lds_off), "v"(gp) : "memory");
        }
    }
    asm volatile("s_wait_asynccnt 0x0" ::: "memory");
    __syncthreads();
#endif

    // Write out [v0 .. v0+255] x [0..31] contiguously: 8192 floats, coalesced.
#pragma unroll
    for (int j = 0; j < 32; ++j) {
        const long long gi = v0 * FV_C + (long long)j * 256 + t;
        if (gi < FV_V * FV_C) {
            const int local = j * 256 + t;         // 0..8191
            const int vi    = local >> 5;          // voxel within tile
            const int c     = local & 31;          // channel
#if FV_USE_TDM_ASM
            fmT[gi] = tile[c * 257 + vi];          // TDM layout: padded channel rows
#else
            fmT[gi] = tile[vi * 33 + c];           // async layout: padded voxel rows
#endif
        }
    }
}

// ---------------------------------------------------------------------------
// Kernel 2: trilinear gather.  One wave32 per point (lane = channel),
// FV_PPW points per wave.  8 coalesced b32 loads (each = one full 128B line
// for the wave) + 7 fma-lerps + 1 NT store per point.  NT store keeps the
// 275MB volume resident in the 192MB L2.
// TRANSPOSED=false falls back to the original [C][V] layout (strided).
// ---------------------------------------------------------------------------
template <bool TRANSPOSED>
__global__ __launch_bounds__(256) void fv_gather(const float* __restrict__ x,
                                                 const float* __restrict__ vol,
                                                 float* __restrict__ out,
                                                 int npts) {
    const int lane = threadIdx.x & 31;
    const int wave = (int)((blockIdx.x * blockDim.x + threadIdx.x) >> 5);
    const int p0   = wave * FV_PPW;
    if (p0 >= npts) return;

    // One coalesced load covers the coords of all FV_PPW points of this wave.
    float cbuf = 0.0f;
    {
        const long long ci = (long long)p0 * 3 + lane;
        if (lane < 3 * FV_PPW && ci < (long long)npts * 3) cbuf = x[ci];
    }

#pragma unroll
    for (int j = 0; j < FV_PPW; ++j) {
        const int p = p0 + j;
        if (p >= npts) break;

        const float cx = __shfl(cbuf, 3 * j + 0, 32);
        const float cy = __shfl(cbuf, 3 * j + 1, 32);
        const float cz = __shfl(cbuf, 3 * j + 2, 32);

        // i = clip((c+1)*0.5*(S-1), 0, S-1);  0.5*(129-1) == 64 exactly.
        const float fx = fminf(fmaxf((cx + 1.0f) * 64.0f, 0.0f), 128.0f);
        const float fy = fminf(fmaxf((cy + 1.0f) * 64.0f, 0.0f), 128.0f);
        const float fz = fminf(fmaxf((cz + 1.0f) * 64.0f, 0.0f), 128.0f);

        const int x0 = (int)fx, y0 = (int)fy, z0 = (int)fz;   // floor (non-negative)
        const float tx = fx - (float)x0;
        const float ty = fy - (float)y0;
        const float tz = fz - (float)z0;
        const int x1 = min(x0 + 1, FV_S - 1);
        const int y1 = min(y0 + 1, FV_S - 1);
        const int z1 = min(z0 + 1, FV_S - 1);

        const long long r00 = ((long long)z0 * FV_S + y0) * FV_S;
        const long long r01 = ((long long)z0 * FV_S + y1) * FV_S;
        const long long r10 = ((long long)z1 * FV_S + y0) * FV_S;
        const long long r11 = ((long long)z1 * FV_S + y1) * FV_S;

        auto ld = [&](long long vox) -> float {
            if (TRANSPOSED) return vol[vox * FV_C + lane];            // 1 line / wave
            else            return vol[(long long)lane * FV_V + vox]; // fallback
        };

        const float a000 = ld(r00 + x0), a001 = ld(r00 + x1);
        const float a010 = ld(r01 + x0), a011 = ld(r01 + x1);
        const float a100 = ld(r10 + x0), a101 = ld(r10 + x1);
        const float a110 = ld(r11 + x0), a111 = ld(r11 + x1);

        // lerp chain: x, then y, then z  (== reference's 8-term weighted sum)
        const float c00 = fmaf(tx, a001 - a000, a000);
        const float c01 = fmaf(tx, a011 - a010, a010);
        const float c10 = fmaf(tx, a101 - a100, a100);
        const float c11 = fmaf(tx, a111 - a110, a110);
        const float d0  = fmaf(ty, c01 - c00, c00);
        const float d1  = fmaf(ty, c11 - c10, c10);
        const float r   = fmaf(tz, d1 - d0, d0);

        __builtin_nontemporal_store(r, &out[(long long)p * FV_C + lane]);
    }
}

// ---------------------------------------------------------------------------
extern "C" void kernel_launch(void* const* d_in, const int* in_sizes, int n_in,
                              void* d_out, int out_size, void* d_ws, size_t ws_size,
                              hipStream_t stream) {
    const float* x  = (const float*)d_in[0];   // [N, 3]
    const float* fm = (const float*)d_in[1];   // [32, 129, 129, 129]
    float* out      = (float*)d_out;           // [N, 32]
    const int npts  = in_sizes[0] / 3;

    const int waves        = (npts + FV_PPW - 1) / FV_PPW;
    const long long thr    = (long long)waves * 32;
    const int gatherBlocks = (int)((thr + 255) / 256);

    const size_t need = (size_t)FV_V * FV_C * sizeof(float);  // ~275 MB
    if (ws_size >= need) {
        float* fmT = (float*)d_ws;
        const int tBlocks = (int)((FV_V + 255) / 256);
        fv_transpose<<<tBlocks, 256, 0, stream>>>(fm, fmT);
        fv_gather<true><<<gatherBlocks, 256, 0, stream>>>(x, fmT, out, npts);
    } else {
        fv_gather<false><<<gatherBlocks, 256, 0, stream>>>(x, fm, out, npts);
    }
}